// AudioVisual_Transformer_with_Adapter_Head_58909771432052
// MI455X (gfx1250) — compile-verified
//
#include <hip/hip_runtime.h>
#include <math.h>

typedef float v2f __attribute__((ext_vector_type(2)));
typedef float v8f __attribute__((ext_vector_type(8)));

#define V8F_ZERO {0.f, 0.f, 0.f, 0.f, 0.f, 0.f, 0.f, 0.f}

// ---------------- WMMA helpers (fp32 16x16x4, full precision) ----------------
__device__ __forceinline__ v8f wmma4(v2f a, v2f b, v8f c) {
  // D = A(16x4 f32) * B(4x16 f32) + C(16x16 f32)
  return __builtin_amdgcn_wmma_f32_16x16x4_f32(false, a, false, b, (short)0, c,
                                               false, false);
}

__device__ __forceinline__ float gelu_exact(float x) {
  return 0.5f * x * (1.0f + erff(x * 0.70710678118654752440f));
}

__device__ __forceinline__ float wave_sum(float v) {
#pragma unroll
  for (int off = 16; off > 0; off >>= 1) v += __shfl_xor(v, off, 32);
  return v;
}

// Fragment mapping (CDNA5 ISA 7.12.2, f32):
//  A 16x4 : lanes l / l+16 hold row M=l; half=lane>>4 selects K pair {0,1}/{2,3}
//  B 4x16 : lane l holds column N=l; half selects K row pair
//  C/D    : VGPR v, lanes 0..15 -> M=v, lanes 16..31 -> M=v+8; N = lane&15

// ---------------- 1. LPF circulant operator init -----------------------------
// y = ifft2( roll(fft2(x), (H/2,W/2)) * mask ) is a fixed complex linear map:
// K[p,q] = phase(q)/(H*W) * sum_{(u,v) in mask} e^{2pi i (u(yp-yq)/H + v(xp-xq)/W)}
// phase(q) = e^{2pi i (H2*yq/H + W2*xq/W)}   (roll folded into source phase)
__global__ void k_init_lpf(float* __restrict__ Kre, float* __restrict__ Kim,
                           int H, int W, float r2) {
  const int HW = H * W;
  const int tot = HW * HW;
  int i = blockIdx.x * blockDim.x + threadIdx.x;
  if (i >= tot) return;
  int p = i / HW, q = i % HW;
  int yp = p / W, xp = p % W, yq = q / W, xq = q % W;
  float cy = (H - 1) * 0.5f, cx = (W - 1) * 0.5f;
  const float TWO_PI = 6.28318530717958647692f;
  float sr = 0.f, si = 0.f;
  for (int u = 0; u < H; ++u) {
    for (int v = 0; v < W; ++v) {
      float du = (float)u - cy, dv = (float)v - cx;
      if (dv * dv + du * du < r2) {
        float ang = TWO_PI * ((float)(u * (yp - yq)) / (float)H +
                              (float)(v * (xp - xq)) / (float)W);
        sr += cosf(ang);
        si += sinf(ang);
      }
    }
  }
  int H2 = H / 2, W2 = W / 2;
  float angq = TWO_PI * ((float)(H2 * yq) / (float)H + (float)(W2 * xq) / (float)W);
  float pr = cosf(angq), pim = sinf(angq);
  float scale = 1.0f / (float)HW;
  Kre[i] = (sr * pr - si * pim) * scale;
  Kim[i] = (sr * pim + si * pr) * scale;
}

// ---------------- 2. conv1x1 (d -> d/4) + GELU  (WMMA GEMM, 4x M-blocked) ----
// o1[b][co][p] = gelu( sum_ci w[co][ci] * tok[b][1+p][ci] );  M=192,N=HW,K=768
// wave computes 64 x 16 outputs: B fragment shared across 4 A fragments.
__global__ void k_conv1(const float* __restrict__ tok, const float* __restrict__ w,
                        float* __restrict__ out, int ntok, int HW, int NT,
                        int tiles) {
  int wave = threadIdx.x >> 5;
  int flat = blockIdx.x * 8 + wave;
  if (flat >= tiles) return;
  int img = flat / (3 * NT);
  int r = flat % (3 * NT);
  int mt4 = r / NT, nt = r % NT;
  int lane = threadIdx.x & 31, half = lane >> 4, l = lane & 15;
  int mrow = mt4 * 64 + l;
  int col = nt * 16 + l;
  int colc = col < HW ? col : HW - 1;
  const float* Ap = w + (size_t)mrow * 768;  // +16*768 per sub-tile
  const float* Bp = tok + ((size_t)img * ntok + 1 + colc) * 768;
  v8f acc0 = V8F_ZERO, acc1 = V8F_ZERO, acc2 = V8F_ZERO, acc3 = V8F_ZERO;
  for (int k0 = 0; k0 < 768; k0 += 4) {
    int ka = k0 + 2 * half;
    v2f b; b.x = Bp[ka]; b.y = Bp[ka + 1];
    v2f a0; a0.x = Ap[ka];             a0.y = Ap[ka + 1];
    v2f a1; a1.x = Ap[ka + 16 * 768];  a1.y = Ap[ka + 16 * 768 + 1];
    v2f a2; a2.x = Ap[ka + 32 * 768];  a2.y = Ap[ka + 32 * 768 + 1];
    v2f a3; a3.x = Ap[ka + 48 * 768];  a3.y = Ap[ka + 48 * 768 + 1];
    acc0 = wmma4(a0, b, acc0);
    acc1 = wmma4(a1, b, acc1);
    acc2 = wmma4(a2, b, acc2);
    acc3 = wmma4(a3, b, acc3);
  }
  if (col < HW) {
    float* ob = out + (size_t)img * 192 * HW + col;
#pragma unroll
    for (int v = 0; v < 8; ++v) {
      int m = mt4 * 64 + v + 8 * half;
      ob[(size_t)m * HW]        = gelu_exact(acc0[v]);
      ob[(size_t)(m + 16) * HW] = gelu_exact(acc1[v]);
      ob[(size_t)(m + 32) * HW] = gelu_exact(acc2[v]);
      ob[(size_t)(m + 48) * HW] = gelu_exact(acc3[v]);
    }
  }
}

// ---------------- 3. conv3x3 (d/4 -> d) + GELU (implicit GEMM, 4x blocked) ---
// M=768, N=HW, K=192*9; tap loop hoisted; B fragment + tap decode shared.
__global__ void k_conv2(const float* __restrict__ o1, const float* __restrict__ w2,
                        float* __restrict__ out, int H, int W, int NT, int tiles) {
  const int HW = H * W;
  int wave = threadIdx.x >> 5;
  int flat = blockIdx.x * 8 + wave;
  if (flat >= tiles) return;
  int img = flat / (12 * NT);
  int r = flat % (12 * NT);
  int mt4 = r / NT, nt = r % NT;
  int lane = threadIdx.x & 31, half = lane >> 4, l = lane & 15;
  int mrow = mt4 * 64 + l;
  int col = nt * 16 + l;
  int colc = col < HW ? col : HW - 1;
  int y = colc / W, x = colc % W;
  const float* Arow = w2 + (size_t)mrow * 1728;
  const float* Ibase = o1 + (size_t)img * 192 * HW;
  v8f acc0 = V8F_ZERO, acc1 = V8F_ZERO, acc2 = V8F_ZERO, acc3 = V8F_ZERO;
  for (int j9 = 0; j9 < 9; ++j9) {
    int dy = j9 / 3 - 1, dx = j9 % 3 - 1;
    int yy = y + dy, xx = x + dx;
    bool ok = (yy >= 0) && (yy < H) && (xx >= 0) && (xx < W);
    int pos = ok ? (yy * W + xx) : 0;
    const float* Bp = Ibase + pos;
    for (int c0 = 0; c0 < 192; c0 += 4) {
      int ka = c0 + 2 * half;
      v2f b;
      b.x = ok ? Bp[(size_t)ka * HW] : 0.f;
      b.y = ok ? Bp[(size_t)(ka + 1) * HW] : 0.f;
      int ia = ka * 9 + j9, ib = (ka + 1) * 9 + j9;
      v2f a0; a0.x = Arow[ia];             a0.y = Arow[ib];
      v2f a1; a1.x = Arow[ia + 16 * 1728]; a1.y = Arow[ib + 16 * 1728];
      v2f a2; a2.x = Arow[ia + 32 * 1728]; a2.y = Arow[ib + 32 * 1728];
      v2f a3; a3.x = Arow[ia + 48 * 1728]; a3.y = Arow[ib + 48 * 1728];
      acc0 = wmma4(a0, b, acc0);
      acc1 = wmma4(a1, b, acc1);
      acc2 = wmma4(a2, b, acc2);
      acc3 = wmma4(a3, b, acc3);
    }
  }
  if (col < HW) {
    float* ob = out + (size_t)img * 768 * HW + col;
#pragma unroll
    for (int v = 0; v < 8; ++v) {
      int m = mt4 * 64 + v + 8 * half;
      ob[(size_t)m * HW]        = gelu_exact(acc0[v]);
      ob[(size_t)(m + 16) * HW] = gelu_exact(acc1[v]);
      ob[(size_t)(m + 32) * HW] = gelu_exact(acc2[v]);
      ob[(size_t)(m + 48) * HW] = gelu_exact(acc3[v]);
    }
  }
}

// ---------------- 4. LPF apply: |K_complex * x| (2x M-blocked, re+im) --------
// rows = B*768 (X row-major, lda=HW); y[row][p] = |sum_q K[p][q] x[row][q]|
__global__ void k_lpf(const float* __restrict__ X, const float* __restrict__ Kre,
                      const float* __restrict__ Kim, float* __restrict__ out,
                      int HW, int NT, int tiles) {
  int wave = threadIdx.x >> 5;
  int flat = blockIdx.x * 8 + wave;
  if (flat >= tiles) return;
  int mt2 = flat / NT, nt = flat % NT;
  int lane = threadIdx.x & 31, half = lane >> 4, l = lane & 15;
  int row = mt2 * 32 + l;
  int col = nt * 16 + l;
  int colc = col < HW ? col : HW - 1;
  const float* Ap0 = X + (size_t)row * HW;
  const float* Ap1 = Ap0 + (size_t)16 * HW;
  const float* Rr = Kre + (size_t)colc * HW;
  const float* Ri = Kim + (size_t)colc * HW;
  v8f ar0 = V8F_ZERO, ai0 = V8F_ZERO, ar1 = V8F_ZERO, ai1 = V8F_ZERO;
  for (int k0 = 0; k0 < HW; k0 += 4) {
    int ka = k0 + 2 * half;
    v2f br; br.x = Rr[ka]; br.y = Rr[ka + 1];
    v2f bi; bi.x = Ri[ka]; bi.y = Ri[ka + 1];
    v2f a0; a0.x = Ap0[ka]; a0.y = Ap0[ka + 1];
    v2f a1; a1.x = Ap1[ka]; a1.y = Ap1[ka + 1];
    ar0 = wmma4(a0, br, ar0);
    ai0 = wmma4(a0, bi, ai0);
    ar1 = wmma4(a1, br, ar1);
    ai1 = wmma4(a1, bi, ai1);
  }
  if (col < HW) {
#pragma unroll
    for (int v = 0; v < 8; ++v) {
      int m = mt2 * 32 + v + 8 * half;
      out[(size_t)m * HW + col] = sqrtf(ar0[v] * ar0[v] + ai0[v] * ai0[v]);
      out[(size_t)(m + 16) * HW + col] =
          sqrtf(ar1[v] * ar1[v] + ai1[v] * ai1[v]);
    }
  }
}

// ---------------- 5. q/k/v projection (WMMA GEMM, 4x blocked, scatter) -------
// qkv[s][b][head][p][c] = sum_i w_s[head*64+c][i] * o3[b][i][p]
__global__ void k_proj(const float* __restrict__ o3, const float* __restrict__ wq,
                       const float* __restrict__ wk, const float* __restrict__ wv,
                       float* __restrict__ outb, int Bimg, int HW, int NT,
                       int tiles) {
  int wave = threadIdx.x >> 5;
  int flat = blockIdx.x * 8 + wave;
  if (flat >= tiles) return;
  int tilesPer = Bimg * 12 * NT;
  int s = flat / tilesPer;
  int r = flat % tilesPer;
  int img = r / (12 * NT);
  int r2 = r % (12 * NT);
  int mt4 = r2 / NT, nt = r2 % NT;
  int lane = threadIdx.x & 31, half = lane >> 4, l = lane & 15;
  int mrow = mt4 * 64 + l;
  int col = nt * 16 + l;
  int colc = col < HW ? col : HW - 1;
  const float* w = (s == 0) ? wq : ((s == 1) ? wk : wv);
  const float* Ap = w + (size_t)mrow * 768;
  const float* Bbase = o3 + (size_t)img * 768 * HW + colc;
  v8f acc0 = V8F_ZERO, acc1 = V8F_ZERO, acc2 = V8F_ZERO, acc3 = V8F_ZERO;
  for (int k0 = 0; k0 < 768; k0 += 4) {
    int ka = k0 + 2 * half;
    v2f b;
    b.x = Bbase[(size_t)ka * HW];
    b.y = Bbase[(size_t)(ka + 1) * HW];
    v2f a0; a0.x = Ap[ka];            a0.y = Ap[ka + 1];
    v2f a1; a1.x = Ap[ka + 16 * 768]; a1.y = Ap[ka + 16 * 768 + 1];
    v2f a2; a2.x = Ap[ka + 32 * 768]; a2.y = Ap[ka + 32 * 768 + 1];
    v2f a3; a3.x = Ap[ka + 48 * 768]; a3.y = Ap[ka + 48 * 768 + 1];
    acc0 = wmma4(a0, b, acc0);
    acc1 = wmma4(a1, b, acc1);
    acc2 = wmma4(a2, b, acc2);
    acc3 = wmma4(a3, b, acc3);
  }
  if (col < HW) {
    size_t base = (((size_t)s * Bimg + img) * 12) * HW * 64 + (size_t)col * 64;
#pragma unroll
    for (int v = 0; v < 8; ++v) {
      int m0 = mt4 * 64 + v + 8 * half;
#pragma unroll
      for (int i = 0; i < 4; ++i) {
        int m = m0 + 16 * i;
        int head = m >> 6, c = m & 63;
        float val = (i == 0) ? acc0[v] : (i == 1) ? acc1[v]
                    : (i == 2) ? acc2[v] : acc3[v];
        outb[base + (size_t)head * HW * 64 + c] = val;
      }
    }
  }
}

// ---------------- 6. enhancer: rep2[b] = cross + softmax(cross.audio).audio --
__global__ void k_rep2(const float* __restrict__ fa, const float* __restrict__ cross,
                       float* __restrict__ rep2) {
  int b = blockIdx.x;  // 8
  __shared__ float sc[4][20];
  __shared__ float ca[4][20];
  int wave = threadIdx.x >> 5, lane = threadIdx.x & 31;
  for (int pi = wave; pi < 80; pi += 8) {
    int t = pi / 20, mm = pi % 20;
    const float* arow = fa + ((size_t)b * 21 + 1 + mm) * 768;
    const float* crow = cross + (size_t)t * 768;
    float s = 0.f;
    for (int c = lane; c < 768; c += 32) s += arow[c] * crow[c];
    s = wave_sum(s);
    if (lane == 0) sc[t][mm] = s;
  }
  __syncthreads();
  if (threadIdx.x < 4) {
    int t = threadIdx.x;
    float mx = -1e30f;
    for (int mm = 0; mm < 20; ++mm) mx = fmaxf(mx, sc[t][mm]);
    float sum = 0.f;
    for (int mm = 0; mm < 20; ++mm) {
      float e = expf(sc[t][mm] - mx);
      ca[t][mm] = e;
      sum += e;
    }
    float inv = 1.0f / sum;
    for (int mm = 0; mm < 20; ++mm) ca[t][mm] *= inv;
  }
  __syncthreads();
  for (int idx = threadIdx.x; idx < 4 * 768; idx += 256) {
    int t = idx / 768, c = idx % 768;
    float v = cross[(size_t)t * 768 + c];
    for (int mm = 0; mm < 20; ++mm)
      v += ca[t][mm] * fa[((size_t)b * 21 + 1 + mm) * 768 + c];
    rep2[((size_t)b * 4 + t) * 768 + c] = v;
  }
}

// ---------------- 7. enhancer rows: cross-attn + residual + LayerNorm --------
// one block per video token row (1576); wave-per-frame, wave32 shfl reductions
__global__ void k_enh(const float* __restrict__ fv, const float* __restrict__ rep2,
                      const float* __restrict__ g, const float* __restrict__ beta,
                      float* __restrict__ xln) {
  int vr = blockIdx.x;          // 0..1575  (= b*197 + m)
  int b = vr / 197, m = vr % 197;
  int wave = threadIdx.x >> 5, lane = threadIdx.x & 31;
  const float* rb = rep2 + (size_t)b * 4 * 768;
  for (int f = wave; f < 16; f += 8) {
    const float* vrow = fv + (((size_t)b * 16 + f) * 197 + m) * 768;
    float vals[24];
    if (m == 0) {
#pragma unroll
      for (int i = 0; i < 24; ++i) vals[i] = vrow[lane + 32 * i];
    } else {
      float s0 = 0.f, s1 = 0.f, s2 = 0.f, s3 = 0.f;
      for (int c = lane; c < 768; c += 32) {
        float v = vrow[c];
        s0 += v * rb[c];
        s1 += v * rb[768 + c];
        s2 += v * rb[1536 + c];
        s3 += v * rb[2304 + c];
      }
      s0 = wave_sum(s0); s1 = wave_sum(s1);
      s2 = wave_sum(s2); s3 = wave_sum(s3);
      float mx = fmaxf(fmaxf(s0, s1), fmaxf(s2, s3));
      float e0 = expf(s0 - mx), e1 = expf(s1 - mx);
      float e2 = expf(s2 - mx), e3 = expf(s3 - mx);
      float inv = 1.0f / (e0 + e1 + e2 + e3);
      e0 *= inv; e1 *= inv; e2 *= inv; e3 *= inv;
#pragma unroll
      for (int i = 0; i < 24; ++i) {
        int c = lane + 32 * i;
        vals[i] = vrow[c] + e0 * rb[c] + e1 * rb[768 + c] + e2 * rb[1536 + c] +
                  e3 * rb[2304 + c];
      }
    }
    float sum = 0.f, sq = 0.f;
#pragma unroll
    for (int i = 0; i < 24; ++i) { sum += vals[i]; sq += vals[i] * vals[i]; }
    sum = wave_sum(sum);
    sq = wave_sum(sq);
    float mu = sum * (1.0f / 768.0f);
    float var = sq * (1.0f / 768.0f) - mu * mu;
    float inv = rsqrtf(var + 1e-5f);
    size_t row = (size_t)vr * 16 + f;
#pragma unroll
    for (int i = 0; i < 24; ++i) {
      int c = lane + 32 * i;
      xln[row * 768 + c] = (vals[i] - mu) * inv * g[c] + beta[c];
    }
  }
}

// -------- 8. MLP GEMM: out = act(X(RxK) * W(NxK)^T + bias), 4x blocked -------
__global__ void k_mlp(const float* __restrict__ X, const float* __restrict__ W,
                      const float* __restrict__ bias, float* __restrict__ out,
                      int K, int N, int NT, int tiles, int doGelu) {
  int wave = threadIdx.x >> 5;
  int flat = blockIdx.x * 8 + wave;
  if (flat >= tiles) return;
  int mt4 = flat / NT, nt = flat % NT;
  int lane = threadIdx.x & 31, half = lane >> 4, l = lane & 15;
  int row = mt4 * 64 + l;
  int col = nt * 16 + l;
  const float* Ap = X + (size_t)row * K;
  const float* Bp = W + (size_t)col * K;
  const size_t s16 = (size_t)16 * K;
  v8f acc0 = V8F_ZERO, acc1 = V8F_ZERO, acc2 = V8F_ZERO, acc3 = V8F_ZERO;
  for (int k0 = 0; k0 < K; k0 += 4) {
    int ka = k0 + 2 * half;
    v2f b; b.x = Bp[ka]; b.y = Bp[ka + 1];
    v2f a0; a0.x = Ap[ka];            a0.y = Ap[ka + 1];
    v2f a1; a1.x = Ap[ka + s16];      a1.y = Ap[ka + s16 + 1];
    v2f a2; a2.x = Ap[ka + 2 * s16];  a2.y = Ap[ka + 2 * s16 + 1];
    v2f a3; a3.x = Ap[ka + 3 * s16];  a3.y = Ap[ka + 3 * s16 + 1];
    acc0 = wmma4(a0, b, acc0);
    acc1 = wmma4(a1, b, acc1);
    acc2 = wmma4(a2, b, acc2);
    acc3 = wmma4(a3, b, acc3);
  }
  float bz = bias[col];
  float* ob = out + col;
#pragma unroll
  for (int v = 0; v < 8; ++v) {
    int m = mt4 * 64 + v + 8 * half;
#pragma unroll
    for (int i = 0; i < 4; ++i) {
      float val = ((i == 0) ? acc0[v] : (i == 1) ? acc1[v]
                   : (i == 2) ? acc2[v] : acc3[v]) + bz;
      if (doGelu) val = gelu_exact(val);
      ob[(size_t)(m + 16 * i) * N] = val;
    }
  }
}

// ---------------- workspace layout (floats) ----------------------------------
static const size_t OFF_KRE_S = 0;                         // 196*196
static const size_t OFF_KIM_S = 38416;                     // 196*196
static const size_t OFF_KRE_A = 76832;                     // 20*20
static const size_t OFF_KIM_A = 77232;                     // 20*20
static const size_t OFF_BUF_A = 77824;                     // o1_s (4.82M) / h1 (4.84M)
static const size_t OFF_BUF_B = 4919296;                   // o2_s (19.27M) / xln (19.37M)
static const size_t OFF_O3S   = 24285184;                  // 19.27M
static const size_t OFF_O1A   = 43552768;                  // 30720
static const size_t OFF_O2A   = 43583488;                  // 122880
static const size_t OFF_O3A   = 43706368;                  // 122880
static const size_t OFF_REP2  = 43829248;                  // 24576

static const size_t AUD_OUT_OFF = 57802752;   // 3*128*12*196*64
static const size_t ENH_OUT_OFF = 58171392;   // + 3*8*12*20*64

extern "C" void kernel_launch(void* const* d_in, const int* in_sizes, int n_in,
                              void* d_out, int out_size, void* d_ws, size_t ws_size,
                              hipStream_t stream) {
  const float* fa      = (const float*)d_in[0];
  const float* fv      = (const float*)d_in[1];
  const float* v_conv1 = (const float*)d_in[4];
  const float* v_conv2 = (const float*)d_in[5];
  const float* v_q     = (const float*)d_in[6];
  const float* v_k     = (const float*)d_in[7];
  const float* v_v     = (const float*)d_in[8];
  const float* a_conv1 = (const float*)d_in[9];
  const float* a_conv2 = (const float*)d_in[10];
  const float* a_q     = (const float*)d_in[11];
  const float* a_k     = (const float*)d_in[12];
  const float* a_v     = (const float*)d_in[13];
  const float* cross   = (const float*)d_in[14];
  const float* w1      = (const float*)d_in[15];
  const float* b1      = (const float*)d_in[16];
  const float* w2      = (const float*)d_in[17];
  const float* b2      = (const float*)d_in[18];
  const float* ln_g    = (const float*)d_in[19];
  const float* ln_b    = (const float*)d_in[20];

  float* ws   = (float*)d_ws;
  float* dout = (float*)d_out;

  float* KreS = ws + OFF_KRE_S;
  float* KimS = ws + OFF_KIM_S;
  float* KreA = ws + OFF_KRE_A;
  float* KimA = ws + OFF_KIM_A;
  float* o1s  = ws + OFF_BUF_A;
  float* o2s  = ws + OFF_BUF_B;
  float* o3s  = ws + OFF_O3S;
  float* o1a  = ws + OFF_O1A;
  float* o2a  = ws + OFF_O2A;
  float* o3a  = ws + OFF_O3A;
  float* rep2 = ws + OFF_REP2;
  float* xln  = ws + OFF_BUF_B;   // reuse o2_s region (dead by then)
  float* h1   = ws + OFF_BUF_A;   // reuse o1_s region (dead by then)

  dim3 blk(256);
  auto nb = [](long tiles) { return (unsigned)((tiles + 7) / 8); };

  // LPF circulant operators
  k_init_lpf<<<(38416 + 255) / 256, blk, 0, stream>>>(KreS, KimS, 14, 14, 3.0f);
  k_init_lpf<<<(400 + 255) / 256, blk, 0, stream>>>(KreA, KimA, 4, 5, 1.0f);

  // ---- spatio adapter (B=128, HW=196, NT=13) ----
  // conv1: tiles = 128 * (192/64) * 13 = 4992
  k_conv1<<<nb(4992), blk, 0, stream>>>(fv, v_conv1, o1s, 197, 196, 13, 4992);
  // conv2: tiles = 128 * (768/64) * 13 = 19968
  k_conv2<<<nb(19968), blk, 0, stream>>>(o1s, v_conv2, o2s, 14, 14, 13, 19968);
  // lpf: tiles = (98304/32) * 13 = 39936
  k_lpf<<<nb(39936), blk, 0, stream>>>(o2s, KreS, KimS, o3s, 196, 13, 39936);
  // proj: tiles = 3 * 128 * 12 * 13 = 59904
  k_proj<<<nb(59904), blk, 0, stream>>>(o3s, v_q, v_k, v_v, dout, 128, 196, 13,
                                        59904);

  // ---- audio adapter (B=8, HW=20, NT=2) ----
  k_conv1<<<nb(48), blk, 0, stream>>>(fa, a_conv1, o1a, 21, 20, 2, 48);
  k_conv2<<<nb(192), blk, 0, stream>>>(o1a, a_conv2, o2a, 4, 5, 2, 192);
  k_lpf<<<nb(384), blk, 0, stream>>>(o2a, KreA, KimA, o3a, 20, 2, 384);
  k_proj<<<nb(576), blk, 0, stream>>>(o3a, a_q, a_k, a_v, dout + AUD_OUT_OFF, 8,
                                      20, 2, 576);

  // ---- enhancer ----
  k_rep2<<<8, blk, 0, stream>>>(fa, cross, rep2);
  k_enh<<<1576, blk, 0, stream>>>(fv, rep2, ln_g, ln_b, xln);
  // h1 = gelu(xln @ w1^T + b1): R=25216 (394 groups of 64), N=192 (NT=12)
  k_mlp<<<nb(4728), blk, 0, stream>>>(xln, w1, b1, h1, 768, 192, 12, 4728, 1);
  // enhanced = h1 @ w2^T + b2: N=768 (NT=48), tiles = 394*48 = 18912
  k_mlp<<<nb(18912), blk, 0, stream>>>(h1, w2, b2, dout + ENH_OUT_OFF, 192, 768,
                                       48, 18912, 0);
}